// torchIWT_74483322847846
// MI455X (gfx1250) — compile-verified
//
#include <hip/hip_runtime.h>

#define NN 4096
#define TH 0.01f
// 1/(2*MM) with MM = 3.2^2 = 10.24 ; 1/20.48 = 0.048828125 exactly
#define INV2MM 0.048828125f

typedef __attribute__((ext_vector_type(2))) float v2f;
typedef __attribute__((ext_vector_type(8))) float v8f;

__device__ __forceinline__ float get_phi(float u) {
    float t = 3.14159265358979323846f * u;
    if (t == 0.0f) return 1.0f;
    return __builtin_sinf(t) * __builtin_expf(-u * u * INV2MM) / t;
}

// For each of K targets t_i = (i-s)/s, find argmin_n |x[n]-t_i| over sorted x.
__global__ void nearest_kernel(const float* __restrict__ x, int n, int s, int K,
                               int* __restrict__ idx) {
    int i = blockIdx.x * blockDim.x + threadIdx.x;
    if (i >= K) return;
    float t = (float)(i - s) / (float)s;
    // lower_bound: first index with x[lo] >= t
    int lo = 0, hi = n;
    while (lo < hi) {
        int mid = (lo + hi) >> 1;
        if (x[mid] < t) lo = mid + 1; else hi = mid;
    }
    int best;
    if (lo == 0) best = 0;
    else if (lo >= n) best = n - 1;
    else {
        float d1 = fabsf(x[lo - 1] - t);
        float d2 = fabsf(x[lo] - t);
        best = (d1 <= d2) ? (lo - 1) : lo;  // tie -> lower index (argmin first-hit)
    }
    idx[i] = best;
}

// psi[n][i] = phi(s*x[n] - (i-s)) for i<K, 0 padding for i in [K,KP)
__global__ void psi_kernel(const float* __restrict__ x, int s, int K, int KP,
                           float* __restrict__ psi) {
    int id = blockIdx.x * blockDim.x + threadIdx.x;
    if (id >= NN * KP) return;
    int n = id / KP;
    int i = id - n * KP;
    float v = 0.0f;
    if (i < K) v = get_phi((float)s * x[n] - (float)(i - s));
    psi[id] = v;
}

// D[a][b] = threshold(img[ic[a],ir[b]] - approx[ic[a],ir[b]]) (or raw img for stage 0)
__global__ void gather_D_kernel(const float* __restrict__ img,
                                const float* __restrict__ approx,
                                const int* __restrict__ ic, const int* __restrict__ ir,
                                int K, int KP, int use_th, float* __restrict__ D) {
    int id = blockIdx.x * blockDim.x + threadIdx.x;
    if (id >= KP * KP) return;
    int a = id / KP;
    int b = id - a * KP;
    float v = 0.0f;
    if (a < K && b < K) {
        long long off = (long long)ic[a] * NN + ir[b];
        float f = img[off];
        if (use_th) {
            float d = f - approx[off];
            v = (fabsf(d) > TH) ? d : 0.0f;
        } else {
            v = f;
        }
    }
    D[id] = v;
}

// T (NN x KP) = psi_c (NN x KP) @ D (KP x KP)   -- small GEMM, plain VALU
__global__ void gemm_T_kernel(const float* __restrict__ psiC,
                              const float* __restrict__ D, int KP,
                              float* __restrict__ T) {
    int col = blockIdx.x * blockDim.x + threadIdx.x;
    int row = blockIdx.y;
    if (col >= KP) return;
    float acc = 0.0f;
    const float* pr = psiC + (long long)row * KP;
    for (int k = 0; k < KP; ++k)
        acc = fmaf(pr[k], D[(long long)k * KP + col], acc);
    T[(long long)row * KP + col] = acc;
}

// approx (NN x NN) (+)= T (NN x KP) @ psi_r^T  (NT GEMM via V_WMMA_F32_16X16X4_F32)
// Block: 256 threads = 8 waves. Each wave: one 32x32 macro-tile (2x2 WMMA tiles).
// Block tile: 64 rows x 128 cols. Grid (4096/128, 4096/64) -> EXEC always all-1s.
__global__ void __launch_bounds__(256)
wmma_update_kernel(const float* __restrict__ T, const float* __restrict__ psiR,
                   float* __restrict__ out, int KP, int accumulate) {
    const int lane = threadIdx.x & 31;
    const int wave = threadIdx.x >> 5;           // 0..7
    const int ll = lane & 15;                     // N / M sub-index
    const int lh = lane >> 4;                     // lane-half
    const int r0 = blockIdx.y * 64 + (wave >> 2) * 32;
    const int c0 = blockIdx.x * 128 + (wave & 3) * 32;

    v8f c00 = {}, c01 = {}, c10 = {}, c11 = {};
    if (accumulate) {
        // C/D layout: vgpr v, lane L -> m = v + 8*(L/16), n = L%16
        for (int v = 0; v < 8; ++v) {
            int mr = v + 8 * lh;
            c00[v] = out[(long long)(r0 + mr) * NN + (c0 + ll)];
            c01[v] = out[(long long)(r0 + mr) * NN + (c0 + 16 + ll)];
            c10[v] = out[(long long)(r0 + 16 + mr) * NN + (c0 + ll)];
            c11[v] = out[(long long)(r0 + 16 + mr) * NN + (c0 + 16 + ll)];
        }
    }

    const float* Ta0 = T + (long long)(r0 + ll) * KP;        // A tile rows
    const float* Ta1 = T + (long long)(r0 + 16 + ll) * KP;
    const float* Pb0 = psiR + (long long)(c0 + ll) * KP;      // B^T rows
    const float* Pb1 = psiR + (long long)(c0 + 16 + ll) * KP;
    const int kb = 2 * lh;  // A/B frag: lane L holds K = 2*(L/16) + {0,1}

    for (int k0 = 0; k0 < KP; k0 += 4) {
        v2f a0 = *(const v2f*)(Ta0 + k0 + kb);
        v2f a1 = *(const v2f*)(Ta1 + k0 + kb);
        v2f b0 = *(const v2f*)(Pb0 + k0 + kb);
        v2f b1 = *(const v2f*)(Pb1 + k0 + kb);
        c00 = __builtin_amdgcn_wmma_f32_16x16x4_f32(false, a0, false, b0, (short)0, c00, false, false);
        c01 = __builtin_amdgcn_wmma_f32_16x16x4_f32(false, a0, false, b1, (short)0, c01, false, false);
        c10 = __builtin_amdgcn_wmma_f32_16x16x4_f32(false, a1, false, b0, (short)0, c10, false, false);
        c11 = __builtin_amdgcn_wmma_f32_16x16x4_f32(false, a1, false, b1, (short)0, c11, false, false);
    }

    for (int v = 0; v < 8; ++v) {
        int mr = v + 8 * lh;
        out[(long long)(r0 + mr) * NN + (c0 + ll)]            = c00[v];
        out[(long long)(r0 + mr) * NN + (c0 + 16 + ll)]       = c01[v];
        out[(long long)(r0 + 16 + mr) * NN + (c0 + ll)]       = c10[v];
        out[(long long)(r0 + 16 + mr) * NN + (c0 + 16 + ll)]  = c11[v];
    }
}

extern "C" void kernel_launch(void* const* d_in, const int* in_sizes, int n_in,
                              void* d_out, int out_size, void* d_ws, size_t ws_size,
                              hipStream_t stream) {
    const float* img = (const float*)d_in[0];
    const float* xc  = (const float*)d_in[1];
    const float* xr  = (const float*)d_in[2];
    float* out = (float*)d_out;

    const int KPmax = 272;
    char* ws = (char*)d_ws;
    size_t off = 0;
    auto wsalloc = [&](size_t bytes) -> void* {
        void* p = ws + off;
        off += (bytes + 255) & ~(size_t)255;
        return p;
    };
    float* psiC = (float*)wsalloc((size_t)NN * KPmax * 4);
    float* psiR = (float*)wsalloc((size_t)NN * KPmax * 4);
    float* Tm   = (float*)wsalloc((size_t)NN * KPmax * 4);
    float* Dm   = (float*)wsalloc((size_t)KPmax * KPmax * 4);
    int*   ic   = (int*)wsalloc(KPmax * 4);
    int*   ir   = (int*)wsalloc(KPmax * 4);

    const int sArr[5] = {8, 16, 32, 64, 128};
    for (int st = 0; st < 5; ++st) {
        const int s = sArr[st];
        const int K = 2 * s + 1;
        const int KP = (K + 3) & ~3;
        const int use_th = (st > 0) ? 1 : 0;

        int nb = (K + 127) / 128;
        nearest_kernel<<<nb, 128, 0, stream>>>(xc, NN, s, K, ic);
        nearest_kernel<<<nb, 128, 0, stream>>>(xr, NN, s, K, ir);

        int total = NN * KP;
        psi_kernel<<<(total + 255) / 256, 256, 0, stream>>>(xc, s, K, KP, psiC);
        psi_kernel<<<(total + 255) / 256, 256, 0, stream>>>(xr, s, K, KP, psiR);

        int totD = KP * KP;
        gather_D_kernel<<<(totD + 255) / 256, 256, 0, stream>>>(img, out, ic, ir, K, KP, use_th, Dm);

        dim3 gT((KP + 63) / 64, NN);
        gemm_T_kernel<<<gT, 64, 0, stream>>>(psiC, Dm, KP, Tm);

        dim3 gU(NN / 128, NN / 64);
        wmma_update_kernel<<<gU, 256, 0, stream>>>(Tm, psiR, out, KP, /*accumulate=*/use_th);
    }
}